// OrthogonalMatrixLoss_BT_37520834298331
// MI455X (gfx1250) — compile-verified
//
#include <hip/hip_runtime.h>

typedef float v2f __attribute__((ext_vector_type(2)));
typedef float v8f __attribute__((ext_vector_type(8)));

#define LAMB     0.1f
#define DDIM     1024
#define KDIM     64
#define NT       64           // 16-wide tiles per dim = DDIM/16
#define PAIRS    2080         // NT*(NT+1)/2
#define NBATCH   64
#define TASKS    (NBATCH * PAIRS)   // 133120
#define BLOCKS   1040
#define WPB      8            // waves per block (256 threads, wave32)

__device__ __forceinline__ int tri_off(int i) {
  // offset of row-tile i into the packed upper triangle: i*(2*NT - i + 1)/2
  return (i * (2 * NT - i + 1)) >> 1;
}

__global__ __launch_bounds__(256) void gram_loss_kernel(const float* __restrict__ x,
                                                        float* __restrict__ partial) {
  const int lane  = threadIdx.x & 31;
  const int wave  = threadIdx.x >> 5;
  const int gwave = blockIdx.x * WPB + wave;
  const int lrow  = lane & 15;   // row within 16-row A/B tile
  const int lhalf = lane >> 4;   // selects K pair {0,1} vs {2,3} within a K=4 chunk

  float acc_sum = 0.0f;

  // TASKS == BLOCKS*WPB*16 exactly -> every wave runs 16 uniform iterations
  for (int t = gwave; t < TASKS; t += BLOCKS * WPB) {
    const int batch = t / PAIRS;
    const int p     = t - batch * PAIRS;

    // decode packed upper-triangular index p -> (ti, tj), tj >= ti
    int ti = (int)((129.0f - sqrtf(16641.0f - 8.0f * (float)p)) * 0.5f);
    if (ti < 0) ti = 0;
    if (ti > NT - 1) ti = NT - 1;
    while (ti + 1 <= NT - 1 && tri_off(ti + 1) <= p) ti++;
    while (tri_off(ti) > p) ti--;
    const int tj = ti + (p - tri_off(ti));

    const float* base = x + (size_t)batch * DDIM * KDIM;
    // A lane gather: A[M=lrow, K=4k+2*lhalf .. +1]; B has the identical pattern
    // for a Gram product since B[k, n] = x[j_base+n, k].
    const float* rowA = base + (size_t)(ti * 16 + lrow) * KDIM + 2 * lhalf;
    const float* rowB = base + (size_t)(tj * 16 + lrow) * KDIM + 2 * lhalf;

    v8f c = {};
#pragma unroll
    for (int kk = 0; kk < 16; ++kk) {
      v2f a = *(const v2f*)(rowA + 4 * kk);
      v2f b = *(const v2f*)(rowB + 4 * kk);
      // D = A(16x4,f32) * B(4x16,f32) + C(16x16,f32)
      c = __builtin_amdgcn_wmma_f32_16x16x4_f32(false, a, false, b,
                                                (short)0, c, false, false);
    }

    // fused reduction of this 16x16 tile of m
    const float w = (ti == tj) ? 1.0f : 2.0f;  // symmetry: count mirrored tile
    float tsum = 0.0f;
#pragma unroll
    for (int v = 0; v < 8; ++v) {
      const float m  = c[v];
      const int   gi = ti * 16 + v + 8 * lhalf;  // C layout: M = v + 8*(lane>>4)
      const int   gj = tj * 16 + lrow;           //           N = lane & 15
      if (gi == gj) {
        const float e = 1.0f - m;   // diag: (1-m)^2, lambda terms cancel
        tsum += e * e;
      } else {
        tsum += LAMB * w * m * m;   // off-diag: lambda * m^2 (x2 if mirrored)
      }
    }
    acc_sum += tsum;
  }

  // wave32 reduction
#pragma unroll
  for (int off = 16; off > 0; off >>= 1)
    acc_sum += __shfl_xor(acc_sum, off, 32);

  __shared__ float smem[WPB];
  if (lane == 0) smem[wave] = acc_sum;
  __syncthreads();
  if (threadIdx.x == 0) {
    float s = 0.0f;
#pragma unroll
    for (int wv = 0; wv < WPB; ++wv) s += smem[wv];
    partial[blockIdx.x] = s;
  }
}

__global__ __launch_bounds__(256) void gram_loss_finalize(const float* __restrict__ partial,
                                                          float* __restrict__ out) {
  __shared__ float smem[256];
  float s = 0.0f;
  for (int i = threadIdx.x; i < BLOCKS; i += 256) s += partial[i];
  smem[threadIdx.x] = s;
  __syncthreads();
  for (int stride = 128; stride > 0; stride >>= 1) {
    if (threadIdx.x < stride) smem[threadIdx.x] += smem[threadIdx.x + stride];
    __syncthreads();
  }
  if (threadIdx.x == 0) out[0] = smem[0] * (1.0f / (float)NBATCH);
}

extern "C" void kernel_launch(void* const* d_in, const int* in_sizes, int n_in,
                              void* d_out, int out_size, void* d_ws, size_t ws_size,
                              hipStream_t stream) {
  (void)in_sizes; (void)n_in; (void)out_size; (void)ws_size;
  const float* x       = (const float*)d_in[0];
  float*       partial = (float*)d_ws;          // BLOCKS floats of scratch
  float*       out     = (float*)d_out;

  gram_loss_kernel<<<BLOCKS, 256, 0, stream>>>(x, partial);
  gram_loss_finalize<<<1, 256, 0, stream>>>(partial, out);
}